// HMM_syllable_23295902613902
// MI455X (gfx1250) — compile-verified
//
#include <hip/hip_runtime.h>

#define N_STATES 6
#define M_SYM 26
#define BATCH 8192
#define T_MAXLEN 512

typedef __attribute__((ext_vector_type(2))) float v2f;
typedef __attribute__((ext_vector_type(8))) float v8f;

// d_ws float layout:
//   [0..127]    A_pad[16][8]  linear transition, rows 8..13 replicate rows 0..5, rest 0
//   [128..335]  E_cols[26][8] linear emission columns, E_cols[m][j] = p(x=m|z=j), j>=6 -> 0
//   [336..343]  priors_pad[8] linear priors, j>=6 -> 0

__global__ void hmm_precompute(const float* __restrict__ up,
                               const float* __restrict__ uA,
                               const float* __restrict__ uE,
                               float* __restrict__ ws) {
  if (threadIdx.x != 0 || blockIdx.x != 0) return;
  float* A_pad  = ws;
  float* E_cols = ws + 128;
  float* pri    = ws + 336;
  for (int i = 0; i < 344; ++i) ws[i] = 0.0f;

  // priors: log_softmax over the 6 states, then exp
  {
    float mx = up[0];
    for (int j = 1; j < N_STATES; ++j) mx = fmaxf(mx, up[j]);
    float s = 0.0f;
    for (int j = 0; j < N_STATES; ++j) s += expf(up[j] - mx);
    float lse = logf(s) + mx;
    for (int j = 0; j < N_STATES; ++j) pri[j] = expf(up[j] - lse);
  }
  // transition: softmax over dim 0 (column-stochastic), uA[j*6+i] = u(z_t=j | z_{t-1}=i)
  for (int i = 0; i < N_STATES; ++i) {
    float mx = uA[i];
    for (int j = 1; j < N_STATES; ++j) mx = fmaxf(mx, uA[j * N_STATES + i]);
    float s = 0.0f;
    for (int j = 0; j < N_STATES; ++j) s += expf(uA[j * N_STATES + i] - mx);
    float lse = logf(s) + mx;
    for (int j = 0; j < N_STATES; ++j) {
      float v = expf(uA[j * N_STATES + i] - lse);
      A_pad[j * 8 + i]       = v;   // rows 0..5  (D regs of lanes 0-15)
      A_pad[(8 + j) * 8 + i] = v;   // rows 8..13 (lanes 16-31) -> alpha replicated in all lanes
    }
  }
  // emission: softmax over symbols per state row, store transposed + padded
  for (int j = 0; j < N_STATES; ++j) {
    float mx = uE[j * M_SYM];
    for (int m = 1; m < M_SYM; ++m) mx = fmaxf(mx, uE[j * M_SYM + m]);
    float s = 0.0f;
    for (int m = 0; m < M_SYM; ++m) s += expf(uE[j * M_SYM + m] - mx);
    float lse = logf(s) + mx;
    for (int m = 0; m < M_SYM; ++m) E_cols[m * 8 + j] = expf(uE[j * M_SYM + m] - lse);
  }
}

// One WMMA timestep WITHOUT normalization/log: alpha_new = E_sym (*) (A @ alpha).
// Records (unnormalized sum, current log-scale base) at t == tlast.
__device__ __forceinline__ void hmm_step(float a[6], float& s_out,
                                         float& rec_s, float& rec_b, float lgs_base,
                                         int t, int tlast, int sym, bool lo,
                                         v2f A03, v2f A47, const float* sE) {
  const float4 e03 = *(const float4*)(sE + sym * 8);
  const float2 e45 = *(const float2*)(sE + sym * 8 + 4);
  // B operand (K x N, K=states): lanes 0-15 give K0,K1,K4,K5 ; lanes 16-31 give K2,K3,K6,K7.
  // Row replication means every lane already holds a[0..5] for its column.
  // K6/K7 multiply all-zero A columns, so any finite value is fine -> no cndmask for B47.
  v2f B03, B47;
  B03[0] = lo ? a[0] : a[2];
  B03[1] = lo ? a[1] : a[3];
  B47[0] = a[4];
  B47[1] = a[5];
  v8f c = {0.f, 0.f, 0.f, 0.f, 0.f, 0.f, 0.f, 0.f};
  v8f d = __builtin_amdgcn_wmma_f32_16x16x4_f32(false, A03, false, B03,
                                                (short)0, c, false, false);
  d = __builtin_amdgcn_wmma_f32_16x16x4_f32(false, A47, false, B47,
                                            (short)0, d, false, false);
  a[0] = d[0] * e03.x;
  a[1] = d[1] * e03.y;
  a[2] = d[2] * e03.z;
  a[3] = d[3] * e03.w;
  a[4] = d[4] * e45.x;
  a[5] = d[5] * e45.y;
  float s = ((a[0] + a[1]) + (a[2] + a[3])) + (a[4] + a[5]);
  const bool hit = (t == tlast);        // branchless: EXEC must stay all-1 for WMMA
  rec_s = hit ? s : rec_s;
  rec_b = hit ? lgs_base : rec_b;
  s_out = s;
}

__launch_bounds__(256)
__global__ void hmm_forward_wmma(const int* __restrict__ x,
                                 const int* __restrict__ Tlen,
                                 const float* __restrict__ ws,
                                 float* __restrict__ out) {
  __shared__ __align__(16) float sE[M_SYM * 8];
  for (int i = threadIdx.x; i < M_SYM * 8; i += blockDim.x) sE[i] = ws[128 + i];
  __syncthreads();

  const int lane = threadIdx.x & 31;
  const bool lo = lane < 16;
  const int wid = (blockIdx.x * blockDim.x + threadIdx.x) >> 5;
  const int col = wid * 16 + (lane & 15);       // 16 batch columns per wave

  // Constant A operand, ISA 16x4 layout: VGPR0 = K0 (lanes0-15)/K2 (lanes16-31), VGPR1 = K1/K3.
  const int Mrow = lane & 15;
  const int k0 = lo ? 0 : 2;
  const int k1 = lo ? 4 : 6;
  v2f A03, A47;
  A03[0] = ws[Mrow * 8 + k0];
  A03[1] = ws[Mrow * 8 + k0 + 1];
  A47[0] = ws[Mrow * 8 + k1];
  A47[1] = ws[Mrow * 8 + k1 + 1];

  float pr[6];
#pragma unroll
  for (int r = 0; r < 6; ++r) pr[r] = ws[336 + r];

  const int tlast = Tlen[col] - 1;
  const int4* xp4 = (const int4*)(x + col * T_MAXLEN);

  float a[6];
  float lgs_base = 0.0f;   // log2 of accumulated rescales (group boundaries)
  float rec_s = 1.0f;      // sum recorded at t == tlast (relative to lgs_base then)
  float rec_b = 0.0f;      // lgs_base recorded at t == tlast
  float s_last;

  {  // group 0: t=0 init from priors + 3 WMMA steps, then one rescale
    int4 xq = xp4[0];
    {
      const int sym = xq.x;
      const float4 e03 = *(const float4*)(sE + sym * 8);
      const float2 e45 = *(const float2*)(sE + sym * 8 + 4);
      a[0] = pr[0] * e03.x; a[1] = pr[1] * e03.y; a[2] = pr[2] * e03.z;
      a[3] = pr[3] * e03.w; a[4] = pr[4] * e45.x; a[5] = pr[5] * e45.y;
      float s = ((a[0] + a[1]) + (a[2] + a[3])) + (a[4] + a[5]);
      const bool hit = (0 == tlast);
      rec_s = hit ? s : rec_s;
      rec_b = hit ? lgs_base : rec_b;
      s_last = s;
    }
    hmm_step(a, s_last, rec_s, rec_b, lgs_base, 1, tlast, xq.y, lo, A03, A47, sE);
    hmm_step(a, s_last, rec_s, rec_b, lgs_base, 2, tlast, xq.z, lo, A03, A47, sE);
    hmm_step(a, s_last, rec_s, rec_b, lgs_base, 3, tlast, xq.w, lo, A03, A47, sE);
    lgs_base += __log2f(s_last);
    const float inv = __builtin_amdgcn_rcpf(s_last);
    a[0] *= inv; a[1] *= inv; a[2] *= inv;
    a[3] *= inv; a[4] *= inv; a[5] *= inv;
  }
  for (int g = 1; g < T_MAXLEN / 4; ++g) {
    int4 xq = xp4[g];
    const int tb = g * 4;
    hmm_step(a, s_last, rec_s, rec_b, lgs_base, tb + 0, tlast, xq.x, lo, A03, A47, sE);
    hmm_step(a, s_last, rec_s, rec_b, lgs_base, tb + 1, tlast, xq.y, lo, A03, A47, sE);
    hmm_step(a, s_last, rec_s, rec_b, lgs_base, tb + 2, tlast, xq.z, lo, A03, A47, sE);
    hmm_step(a, s_last, rec_s, rec_b, lgs_base, tb + 3, tlast, xq.w, lo, A03, A47, sE);
    // rescale once per 4 steps: alpha normalized to sum 1, fold log2(s) into base.
    // Emission factors >= ~1e-5/step keep 4 unnormalized steps far above f32 denormals.
    lgs_base += __log2f(s_last);
    const float inv = __builtin_amdgcn_rcpf(s_last);
    a[0] *= inv; a[1] *= inv; a[2] *= inv;
    a[3] *= inv; a[4] *= inv; a[5] *= inv;
  }

  // log p(x) = (rec_b + log2(rec_s)) * ln2
  if (lo) out[col] = (rec_b + __log2f(rec_s)) * 0.6931471805599453f;
}

extern "C" void kernel_launch(void* const* d_in, const int* in_sizes, int n_in,
                              void* d_out, int out_size, void* d_ws, size_t ws_size,
                              hipStream_t stream) {
  const int* x    = (const int*)d_in[0];
  const int* T    = (const int*)d_in[1];
  const float* up = (const float*)d_in[2];
  const float* uA = (const float*)d_in[3];
  const float* uE = (const float*)d_in[4];
  float* out = (float*)d_out;
  float* ws  = (float*)d_ws;

  hmm_precompute<<<1, 1, 0, stream>>>(up, uA, uE, ws);

  // 8192 batch / 16 cols-per-wave = 512 waves; 256 threads = 8 waves per block.
  const int threads = 256;
  const int waves = BATCH / 16;
  const int blocks = (waves * 32) / threads;  // 64
  hmm_forward_wmma<<<blocks, threads, 0, stream>>>(x, T, ws, out);
}